// ElectrostaticEnergy_81733227643306
// MI455X (gfx1250) — compile-verified
//
#include <hip/hip_runtime.h>
#include <hip/hip_bf16.h>
#include <stdint.h>

// ---------------------------------------------------------------------------
// ElectrostaticEnergy (PhysNet damped/shielded Coulomb) for MI455X / gfx1250.
//
// Memory-bound: ~128MB sequential edge stream + L2-resident gathers/atomics.
// Strategy: TDM (tensor_load_to_lds) double-buffered staging of the edge
// stream into LDS; wave32 compute from LDS; fp32 HW atomics for the two
// segment sums. No matrix structure -> no WMMA (tensor ops are the relevant
// CDNA5 path here).
// ---------------------------------------------------------------------------

#define TILE_EDGES 1024
#define BLOCK 256
#define EDGES_PER_THREAD (TILE_EDGES / BLOCK)
#define MAX_BLOCKS 1024

typedef unsigned int v4u __attribute__((ext_vector_type(4)));
typedef int          v8i __attribute__((ext_vector_type(8)));
typedef int          v4i __attribute__((ext_vector_type(4)));

// ---- physics constants (f32; derived from the reference's doubles) --------
// CUTON=2.5, CUTOFF=7.5, LR_CUTOFF=10
// CUTON16 = 2.5^16
// CUT_RCONSTANT = 10^15 / (10^16 + 2.5^16)^(17/16)  = 9.999997671684e-3
// CUT_CONSTANT  = (2.5^16+10^16)^(-1/16) + 10^16/(10^16+2.5^16)^(17/16)
//               = 0.1999999767153890
#define KEHALF_F      0.5f
#define CUTON_F       2.5f
#define CUTOFF_F      7.5f
#define INV_RANGE_F   0.2f        /* 1/(CUTOFF-CUTON) */
#define CUTON16_F     2328306.436538696f
#define LR_F          10.0f
#define INV_LR2_F     0.01f
#define TWO_OVER_LR_F 0.2f
#define CUT_RCONST_F  9.999997671684e-3f
#define CUT_CONST_F   0.1999999767153890f

// ---------------------------------------------------------------------------
// 1-D TDM copy: nelem dwords from global -> LDS (descriptor per ISA §8).
// All inputs are workgroup-uniform. Issued by one wave only.
// clang-23 toolchain: 6-arg builtin (uint32x4, int32x8, int32x4, int32x4,
// int32x8, i32 cpol); trailing int32x8 passed zero-filled.
// ---------------------------------------------------------------------------
__device__ __forceinline__ void tdm_load_1d(uint32_t lds_byte_addr,
                                            const void* gptr,
                                            uint32_t nelem /* dwords, <=65535 */) {
  uint64_t ga = (uint64_t)(uintptr_t)gptr;

  v4u g0;
  g0.x = 1u;                                             // count=1, user mode
  g0.y = lds_byte_addr;                                  // LDS byte address
  g0.z = (uint32_t)(ga & 0xFFFFFFFFu);                   // global_addr[31:0]
  g0.w = (uint32_t)((ga >> 32) & 0x01FFFFFFu)            // global_addr[56:32]
         | (2u << 30);                                   // type = 2 ("image")

  v8i g1;
  g1[0] = (int)(2u << 16);                               // data_size=4B; mask=0
  g1[1] = (int)((nelem & 0xFFFFu) << 16);                // tensor_dim0[15:0]
  g1[2] = (int)(((nelem >> 16) & 0xFFFFu) | (1u << 16)); // td0[31:16] | tensor_dim1=1
  g1[3] = (int)((nelem & 0xFFFFu) << 16);                // td1 hi=0 | tile_dim0=nelem
  g1[4] = 1;                                             // tile_dim1=1, tile_dim2=0
  g1[5] = (int)nelem;                                    // tensor_dim0_stride lo
  g1[6] = 0;                                             // stride hi / td1_stride lo
  g1[7] = 0;

  v4i g2; g2[0] = 1; g2[1] = 1; g2[2] = 0; g2[3] = 0;    // tensor_dim2=1, tensor_dim3=1
  v4i g3; g3[0] = 0; g3[1] = 0x00010000; g3[2] = 0; g3[3] = 0; // tensor_dim4=1

  v8i gx = {0, 0, 0, 0, 0, 0, 0, 0};                     // unused trailing group

  __builtin_amdgcn_tensor_load_to_lds(g0, g1, g2, g3, gx, /*cpol=*/0);
}

__device__ __forceinline__ void issue_tile_loads(const float* __restrict__ r_ij,
                                                 const int* __restrict__ idx_i,
                                                 const int* __restrict__ idx_j,
                                                 float* s_r, int* s_ii, int* s_jj,
                                                 int start, int cnt) {
  tdm_load_1d((uint32_t)(uintptr_t)(void*)s_r,
              r_ij + (size_t)start * 3, (uint32_t)(cnt * 3));
  tdm_load_1d((uint32_t)(uintptr_t)(void*)s_ii, idx_i + start, (uint32_t)cnt);
  tdm_load_1d((uint32_t)(uintptr_t)(void*)s_jj, idx_j + start, (uint32_t)cnt);
}

// ---------------------------------------------------------------------------
__global__ void ee_zero_kernel(float* __restrict__ e_atom, int n_atoms,
                               float* __restrict__ e_mol, int n_mol) {
  int i = blockIdx.x * blockDim.x + threadIdx.x;
  if (i < n_atoms) e_atom[i] = 0.0f;
  if (i < n_mol)   e_mol[i]  = 0.0f;
}

// ---------------------------------------------------------------------------
__global__ __launch_bounds__(BLOCK)
void ee_edge_kernel(const float* __restrict__ q,
                    const float* __restrict__ r_ij,
                    const int*   __restrict__ idx_i,
                    const int*   __restrict__ idx_j,
                    float*       __restrict__ e_atom,
                    int n_edges, int n_tiles) {
  __shared__ float s_r [2][3 * TILE_EDGES];
  __shared__ int   s_ii[2][TILE_EDGES];
  __shared__ int   s_jj[2][TILE_EDGES];

  const int tid = threadIdx.x;
  const int nb  = gridDim.x;
  const int t0  = blockIdx.x;
  if (t0 >= n_tiles) return;

  const bool issuer = (tid < 32);   // wave 0 only (uniform branch per wave)

  // Prologue: stage first tile into buffer 0.
  if (issuer) {
    int start = t0 * TILE_EDGES;
    int cnt = n_edges - start; if (cnt > TILE_EDGES) cnt = TILE_EDGES;
    issue_tile_loads(r_ij, idx_i, idx_j, &s_r[0][0], &s_ii[0][0], &s_jj[0][0],
                     start, cnt);
  }

  int buf = 0;
  for (int t = t0; t < n_tiles; t += nb) {
    const int tn = t + nb;
    if (issuer) {
      if (tn < n_tiles) {
        int start = tn * TILE_EDGES;
        int cnt = n_edges - start; if (cnt > TILE_EDGES) cnt = TILE_EDGES;
        issue_tile_loads(r_ij, idx_i, idx_j,
                         &s_r[buf ^ 1][0], &s_ii[buf ^ 1][0], &s_jj[buf ^ 1][0],
                         start, cnt);
        __builtin_amdgcn_s_wait_tensorcnt(3);  // 3 in flight for tn -> tile t done
      } else {
        __builtin_amdgcn_s_wait_tensorcnt(0);  // drain: tile t done
      }
    }
    __syncthreads();   // tile t visible to all waves

    const int base = t * TILE_EDGES;
#pragma unroll
    for (int k = 0; k < EDGES_PER_THREAD; ++k) {
      const int e    = tid + k * BLOCK;
      const int edge = base + e;
      if (edge < n_edges) {
        const float x = s_r[buf][3 * e + 0];
        const float y = s_r[buf][3 * e + 1];
        const float z = s_r[buf][3 * e + 2];
        const int ii = s_ii[buf][e];
        const int jj = s_jj[buf][e];

        const float d2    = fmaf(x, x, fmaf(y, y, z * z));
        const float d     = __fsqrt_rn(d2);
        const float inv_d = __frcp_rn(d);

        // quintic switch
        const float u  = (d - CUTON_F) * INV_RANGE_F;
        const float u2 = u * u;
        const float u3 = u2 * u;
        const float u4 = u3 * u;
        const float u5 = u4 * u;
        const float sw = 1.0f - 6.0f * u5 + 15.0f * u4 - 10.0f * u3;
        const float f  = (d < CUTON_F) ? 1.0f : ((d < CUTOFF_F) ? sw : 0.0f);
        const float omf = 1.0f - f;

        // long-range shifted Coulomb
        const float coulomb =
            (d < LR_F) ? (fmaf(d, INV_LR2_F, inv_d) - TWO_OVER_LR_F) : 0.0f;

        // damped short-range: (d^16 + 2.5^16)^(-1/16) via 4x sqrt + rcp
        const float d4  = d2 * d2;
        const float d8  = d4 * d4;
        const float d16 = d8 * d8;
        const float tt  = d16 + CUTON16_F;
        const float t16 = __fsqrt_rn(__fsqrt_rn(__fsqrt_rn(__fsqrt_rn(tt))));
        const float damped = __frcp_rn(t16) + omf * CUT_RCONST_F * d - CUT_CONST_F;

        const float fac = KEHALF_F * q[ii] * q[jj];
        const float ee  = fac * (f * damped + omf * coulomb);

        unsafeAtomicAdd(&e_atom[ii], ee);  // global_atomic_add_f32 (L2-resident)
      }
    }
    __syncthreads();   // all reads of buf done before it is overwritten
    buf ^= 1;
  }
}

// ---------------------------------------------------------------------------
__global__ void ee_mol_kernel(const float* __restrict__ e_atom,
                              const int*   __restrict__ idx_m,
                              float*       __restrict__ e_mol,
                              int n_atoms) {
  int i = blockIdx.x * blockDim.x + threadIdx.x;
  if (i < n_atoms) {
    unsafeAtomicAdd(&e_mol[idx_m[i]], e_atom[i]);
  }
}

// ---------------------------------------------------------------------------
extern "C" void kernel_launch(void* const* d_in, const int* in_sizes, int n_in,
                              void* d_out, int out_size, void* d_ws, size_t ws_size,
                              hipStream_t stream) {
  // setup_inputs order: atomic_numbers, q, r_ij, idx_i, idx_j, idx_m
  const float* q     = (const float*)d_in[1];
  const float* r_ij  = (const float*)d_in[2];
  const int*   idx_i = (const int*)d_in[3];
  const int*   idx_j = (const int*)d_in[4];
  const int*   idx_m = (const int*)d_in[5];

  const int n_atoms = in_sizes[1];
  const int n_edges = in_sizes[3];
  const int n_mol   = out_size;

  float* e_mol  = (float*)d_out;
  float* e_atom = (float*)d_ws;   // n_atoms floats of scratch (800 KB)

  // 1) zero accumulators (must run every call; we accumulate with atomics)
  {
    int n = (n_atoms > n_mol) ? n_atoms : n_mol;
    ee_zero_kernel<<<(n + 255) / 256, 256, 0, stream>>>(e_atom, n_atoms,
                                                        e_mol, n_mol);
  }

  // 2) per-edge energy + edge->atom scatter
  {
    const int n_tiles = (n_edges + TILE_EDGES - 1) / TILE_EDGES;
    const int nb      = (n_tiles < MAX_BLOCKS) ? n_tiles : MAX_BLOCKS;
    ee_edge_kernel<<<nb, BLOCK, 0, stream>>>(q, r_ij, idx_i, idx_j,
                                             e_atom, n_edges, n_tiles);
  }

  // 3) atom -> molecule scatter
  ee_mol_kernel<<<(n_atoms + 255) / 256, 256, 0, stream>>>(e_atom, idx_m,
                                                           e_mol, n_atoms);
}